// InvLosslessPool_32804960207047
// MI455X (gfx1250) — compile-verified
//
#include <hip/hip_runtime.h>
#include <hip/hip_bf16.h>

// depth_to_space (DCR, block=2): in [32,64,64,256] f32 -> out [32,128,128,64] f32
// Pure permutation: 0 FLOPs, 256 MiB total traffic -> HBM-bound (~11.5 us @ 23.3 TB/s).
// Strategy: one wave32 moves one 512-byte half-cell; src is fully sequential,
// dst is a contiguous 512-byte segment. All accesses are b128 with NT hints
// (footprint 256 MiB > 192 MB L2, single-touch streaming).

typedef float v4f __attribute__((ext_vector_type(4)));

namespace {
constexpr int kB = 32, kH = 64, kW = 64, kC = 256;   // input shape
constexpr long long kNFloat = (long long)kB * kH * kW * kC;   // 33,554,432
constexpr int kN4 = (int)(kNFloat / 4);              // 8,388,608 float4 elements
constexpr int kPerThread = 4;                        // independent b128 ops in flight
constexpr int kThreads = 256;                        // 8 wave32s / block
constexpr int kTotalThreads = kN4 / kPerThread;      // 2,097,152
constexpr int kBlocks = kTotalThreads / kThreads;    // 8192
// Output: [32, 128, 128, 64] -> 2048 float4 per output row (128*64/4).
constexpr int kOutRow4 = 2048;
}

__global__ __launch_bounds__(kThreads) void
d2s_stream_kernel(const v4f* __restrict__ in, v4f* __restrict__ out) {
    const int t0 = blockIdx.x * kThreads + threadIdx.x;

#pragma unroll
    for (int k = 0; k < kPerThread; ++k) {
        const int t = t0 + k * kTotalThreads;        // source float4 index (sequential)

        // Streaming non-temporal 128-bit load: global_load_b128 th:NT
        v4f v = __builtin_nontemporal_load(&in[t]);

        // Decode (b,h,w,i,lane) from the flat float4 index.
        const int lane = t & 31;        // float4 within the 512B half-cell
        const int g    = t >> 5;        // half-cell id = ((b*64+h)*64+w)*2 + i
        const int i    = g & 1;
        const int w    = (g >> 1) & 63;
        const int h    = (g >> 7) & 63;
        const int b    = g >> 13;

        // Destination: output row (b*128 + 2h + i), columns [2w, 2w+2), all 64 channels
        // = contiguous 512 bytes starting at float4 index below.
        const int d = (b * 128 + 2 * h + i) * kOutRow4 + w * 32 + lane;

        // Streaming non-temporal 128-bit store: global_store_b128 th:NT
        __builtin_nontemporal_store(v, &out[d]);
    }
}

extern "C" void kernel_launch(void* const* d_in, const int* in_sizes, int n_in,
                              void* d_out, int out_size, void* d_ws, size_t ws_size,
                              hipStream_t stream) {
    (void)in_sizes; (void)n_in; (void)out_size; (void)d_ws; (void)ws_size;
    const v4f* in  = (const v4f*)d_in[0];
    v4f*       out = (v4f*)d_out;
    d2s_stream_kernel<<<kBlocks, kThreads, 0, stream>>>(in, out);
}